// WeightedAggEdge_38611755991048
// MI455X (gfx1250) — compile-verified
//
#include <hip/hip_runtime.h>

#define N_NODES  50000
#define N_EDGES  800000
#define NODE_DIM 128
#define LBL_DIM  64
#define F_OUT    128

typedef float v2f __attribute__((ext_vector_type(2)));
typedef float v8f __attribute__((ext_vector_type(8)));

// ---- monotone order-preserving float <-> uint key (deterministic segment max) ----
__device__ __forceinline__ unsigned f2key(float f) {
    unsigned u = __float_as_uint(f);
    return (u & 0x80000000u) ? ~u : (u | 0x80000000u);
}
__device__ __forceinline__ float key2f(unsigned k) {
    return (k & 0x80000000u) ? __uint_as_float(k & 0x7FFFFFFFu) : __uint_as_float(~k);
}

__device__ __forceinline__ float dot4(float4 a, float4 b) {
    return fmaf(a.x, b.x, fmaf(a.y, b.y, fmaf(a.z, b.z, a.w * b.w)));
}

// ---- CDNA5 fp32 WMMA: D = A(16x4) * B(4x16) + C(16x16) ----
__device__ __forceinline__ v8f wmma_f32_16x16x4(v2f a, v2f b, v8f c) {
#if __has_builtin(__builtin_amdgcn_wmma_f32_16x16x4_f32)
    return __builtin_amdgcn_wmma_f32_16x16x4_f32(false, a, false, b, (short)0, c, false, false);
#else
    v8f d;
    asm volatile("v_wmma_f32_16x16x4_f32 %0, %1, %2, %3"
                 : "=v"(d) : "v"(a), "v"(b), "v"(c));
    return d;
#endif
}

// K_v: v[d] = sum_c W_fc[d][c] * W_attn[c]   (64 threads, 1 block)
__global__ void kv_fold(const float* __restrict__ wfc, const float* __restrict__ wattn,
                        float* __restrict__ vvec) {
    int d = threadIdx.x;                         // 0..63
    const float4* r = (const float4*)(wfc + (size_t)d * F_OUT);
    const float4* w = (const float4*)wattn;      // first 128 entries pair with e_w
    float s = 0.f;
#pragma unroll
    for (int i = 0; i < F_OUT / 4; ++i) s += dot4(r[i], w[i]);
    vvec[d] = s;
}

// K_init: mkey = 0 (== -inf key), den = 0
__global__ void kinit(unsigned* __restrict__ mkey, float* __restrict__ den) {
    int i = blockIdx.x * blockDim.x + threadIdx.x;
    if (i < N_NODES) { mkey[i] = 0u; den[i] = 0.f; }
}

// K1: per edge attention score + segmented max (uint-key atomicMax: deterministic)
__global__ __launch_bounds__(256)
void k1_score(const float* __restrict__ labels, const float* __restrict__ h,
              const int* __restrict__ src, const float* __restrict__ vvec,
              const float* __restrict__ wattn, float* __restrict__ e_arr,
              unsigned* __restrict__ mkey) {
    __shared__ __align__(16) float sv[LBL_DIM];
    __shared__ __align__(16) float swa[NODE_DIM];
    int t = threadIdx.x;
    if (t < LBL_DIM)               sv[t]       = vvec[t];
    if (t >= 64 && t < 64 + NODE_DIM) swa[t - 64] = wattn[F_OUT + (t - 64)];
    __syncthreads();

    int e = blockIdx.x * 256 + t;                // E divisible by 256, no guard
    const float4* lp = (const float4*)(labels + (size_t)e * LBL_DIM);
    const float4* sv4 = (const float4*)sv;
    float a = 0.f;
#pragma unroll
    for (int i = 0; i < LBL_DIM / 4; ++i) a += dot4(lp[i], sv4[i]);

    int s = src[e];
    const float4* hp = (const float4*)(h + (size_t)s * NODE_DIM);
    const float4* wa4 = (const float4*)swa;
#pragma unroll 8
    for (int i = 0; i < NODE_DIM / 4; ++i) a += dot4(hp[i], wa4[i]);

    float ee = a > 0.f ? a : 0.01f * a;          // leaky_relu(0.01)
    e_arr[e] = ee;
    atomicMax(&mkey[s], f2key(ee));
}

// K2: ex = exp(e - m[src]); segmented sum
__global__ __launch_bounds__(256)
void k2_expsum(const int* __restrict__ src, const unsigned* __restrict__ mkey,
               float* __restrict__ e_arr, float* __restrict__ den) {
    int e = blockIdx.x * 256 + threadIdx.x;
    int s = src[e];
    float ex = __expf(e_arr[e] - key2f(mkey[s]));
    e_arr[e] = ex;
    atomicAdd(&den[s], ex);
}

// K2b: gamma = ex / den[src]
__global__ __launch_bounds__(256)
void k2b_gamma(const int* __restrict__ src, const float* __restrict__ den,
               float* __restrict__ e_arr) {
    int e = blockIdx.x * 256 + threadIdx.x;
    e_arr[e] = e_arr[e] / den[src[e]];
}

// K3: recompute e_w tile (16 edges x 128) with V_WMMA_F32_16X16X4_F32, scale by gamma, store.
// One wave per 16-edge tile. 50000 tiles = 6250 blocks * 8 waves -> all waves full (EXEC all-1s).
__global__ __launch_bounds__(256)
void k3_gemm_scale(const float* __restrict__ labels, const float* __restrict__ wfc,
                   const float* __restrict__ gamma, float* __restrict__ out) {
    const int wave = (blockIdx.x * 256 + threadIdx.x) >> 5;   // tile id
    const int lane = threadIdx.x & 31;
    const int half = lane >> 4;                               // 0: K=0,1  1: K=2,3
    const int ln   = lane & 15;
    const int base = wave * 16;                               // first edge of tile

    // A: 16x4 fp32 fragment per ISA layout: lane(ln) = row M, VGPR0/1 = K, K+1
    const float* arow = labels + (size_t)(base + ln) * LBL_DIM + 2 * half;

    v8f acc[8];
    const v8f zero = {0.f, 0.f, 0.f, 0.f, 0.f, 0.f, 0.f, 0.f};
#pragma unroll
    for (int j = 0; j < 8; ++j) acc[j] = zero;

    for (int kc = 0; kc < LBL_DIM / 4; ++kc) {                // 16 K-chunks of 4
        v2f a = *(const v2f*)(arow + 4 * kc);
        const int k0 = 4 * kc + 2 * half;
        const float* bp = wfc + (size_t)k0 * F_OUT + ln;      // B: K=k0,k0+1 rows, col 16j+ln
#pragma unroll
        for (int j = 0; j < 8; ++j) {                         // 8 N-tiles of 16
            v2f b;
            b.x = bp[16 * j];
            b.y = bp[16 * j + F_OUT];
            acc[j] = wmma_f32_16x16x4(a, b, acc[j]);
        }
    }

    // C/D layout: VGPR v -> row M = v + 8*half, col N = 16j + ln
    float gm[8];
#pragma unroll
    for (int v = 0; v < 8; ++v) gm[v] = gamma[base + 8 * half + v];

#pragma unroll
    for (int j = 0; j < 8; ++j) {
#pragma unroll
        for (int v = 0; v < 8; ++v) {
            out[(size_t)(base + 8 * half + v) * F_OUT + 16 * j + ln] = acc[j][v] * gm[v];
        }
    }
}

extern "C" void kernel_launch(void* const* d_in, const int* in_sizes, int n_in,
                              void* d_out, int out_size, void* d_ws, size_t ws_size,
                              hipStream_t stream) {
    const float* h      = (const float*)d_in[0];
    const float* labels = (const float*)d_in[1];
    const float* wfc    = (const float*)d_in[2];
    const float* wattn  = (const float*)d_in[3];
    const int*   src    = (const int*)d_in[4];
    float* out = (float*)d_out;

    // workspace layout (floats): [e/ex/gamma: E][mkey: N][den: N][v: 64]  ~3.6 MB
    float*    ws    = (float*)d_ws;
    float*    e_arr = ws;
    unsigned* mkey  = (unsigned*)(ws + N_EDGES);
    float*    den   = ws + N_EDGES + N_NODES;
    float*    vvec  = ws + N_EDGES + 2 * N_NODES;

    kv_fold<<<1, 64, 0, stream>>>(wfc, wattn, vvec);
    kinit<<<(N_NODES + 255) / 256, 256, 0, stream>>>(mkey, den);
    k1_score<<<N_EDGES / 256, 256, 0, stream>>>(labels, h, src, vvec, wattn, e_arr, mkey);
    k2_expsum<<<N_EDGES / 256, 256, 0, stream>>>(src, mkey, e_arr, den);
    k2b_gamma<<<N_EDGES / 256, 256, 0, stream>>>(src, den, e_arr);
    k3_gemm_scale<<<(N_EDGES / 16) / 8, 256, 0, stream>>>(labels, wfc, e_arr, out);
}